// STFT_Process_85925115724233
// MI455X (gfx1250) — compile-verified
//
#include <hip/hip_runtime.h>

#define N_FFT   4096
#define HOP     1024
#define HALF    2048
#define NBINS   2049
#define SIGLEN  441000
#define NFRAMES 431
#define NBATCH  32

#define BM 64      // k-bins per workgroup tile
#define BN 32      // frames per workgroup tile (per batch)
#define BB 2       // batches per workgroup

typedef __attribute__((ext_vector_type(2))) float v2f;
typedef __attribute__((ext_vector_type(4))) float v4f;
typedef __attribute__((ext_vector_type(8))) float v8f;
typedef int vi4 __attribute__((vector_size(16)));   // matches builtin param type

__device__ __forceinline__ int reflect_idx(int t) {
  t = (t < 0) ? -t : t;
  t = (t >= SIGLEN) ? (2 * SIGLEN - 2 - t) : t;
  return t;
}

#if __has_builtin(__builtin_amdgcn_global_load_async_to_lds_b128)
#define HAVE_ASYNC 1
#else
#define HAVE_ASYNC 0
#endif

#if HAVE_ASYNC
// ===========================================================================
// Async double-buffered version: global_load_async_to_lds_b128 + ASYNCcnt.
// ===========================================================================
#define KC     32                 // samples per chunk
#define LDST   36                 // LDS row stride (144B rows: 16B aligned, conflict-free)
#define CSZ    (BM * LDST)        // 2304 floats per tile (cos == sin == frames size)
#define SINOFF CSZ
#define FRMOFF (2 * CSZ)
#define BUFSZ  (3 * CSZ)          // 6912 floats per buffer
#define NC     (N_FFT / KC)       // 128 chunks

#if __has_builtin(__builtin_amdgcn_s_wait_asynccnt)
#define WAIT_ASYNC(n) __builtin_amdgcn_s_wait_asynccnt(n)
#else
#define WAIT_ASYNC(n) asm volatile("s_wait_asynccnt " #n ::: "memory")
#endif

__device__ __forceinline__ void async_b128(const float* g, float* l) {
  __builtin_amdgcn_global_load_async_to_lds_b128(
      (__attribute__((address_space(1))) vi4*)(void*)g,
      (__attribute__((address_space(3))) vi4*)(void*)l, 0, 0);
}

// Issue async copies of the cos/sin tiles (4 per thread). Always async.
__device__ __forceinline__ void issue_kern(const float* cosk, const float* sink,
                                           float* B, int tid, int kTile, int n0) {
#pragma unroll
  for (int it = 0; it < 2; ++it) {
    int i   = tid + it * 256;                 // 0..511 v4f units (64 rows x 8)
    int row = i >> 3;
    int j   = (i & 7) << 2;
    int kIdx = kTile + row;
    kIdx = (kIdx > NBINS - 1) ? (NBINS - 1) : kIdx;
    const size_t go = (size_t)kIdx * N_FFT + (size_t)(n0 + j);
    async_b128(&cosk[go], &B[row * LDST + j]);
    async_b128(&sink[go], &B[SINOFF + row * LDST + j]);
  }
}

// Interior blocks: frames need no reflect/clamp -> pure async (2 per thread).
__device__ __forceinline__ void issue_frm(const float* x, float* B, int tid,
                                          int fTile, int bTile, int n0) {
#pragma unroll
  for (int it = 0; it < 2; ++it) {
    int i   = tid + it * 256;                 // 0..511 v4f units
    int bb  = i >> 8;
    int rem = i & 255;
    int f   = rem >> 3;
    int j   = (rem & 7) << 2;
    int t   = (fTile + f) * HOP + n0 + j - HALF;   // interior: 0 <= t, t+3 < SIGLEN
    async_b128(&x[(size_t)(bTile + bb) * SIGLEN + t],
               &B[FRMOFF + (bb * BN + f) * LDST + j]);
  }
}

// Edge blocks: frames via regs with reflect/clamp (2 v4f per thread).
__device__ __forceinline__ void load_frm_edge(const float* x, v4f fr[2], int tid,
                                              int fTile, int bTile, int n0) {
#pragma unroll
  for (int it = 0; it < 2; ++it) {
    int i   = tid + it * 256;
    int bb  = i >> 8;
    int rem = i & 255;
    int f   = rem >> 3;
    int j   = (rem & 7) << 2;
    int fi  = fTile + f;
    fi = (fi > NFRAMES - 1) ? (NFRAMES - 1) : fi;
    int t = fi * HOP + n0 + j - HALF;
    const float* xb = x + (size_t)(bTile + bb) * SIGLEN;
    v4f v;
    if (t >= 0 && t + 3 < SIGLEN) {
      v = *(const v4f*)&xb[t];
    } else {
      v.x = xb[reflect_idx(t)];
      v.y = xb[reflect_idx(t + 1)];
      v.z = xb[reflect_idx(t + 2)];
      v.w = xb[reflect_idx(t + 3)];
    }
    fr[it] = v;
  }
}

__device__ __forceinline__ void store_frm_edge(float* B, const v4f fr[2], int tid) {
#pragma unroll
  for (int it = 0; it < 2; ++it) {
    int i   = tid + it * 256;
    int bb  = i >> 8;
    int rem = i & 255;
    int f   = rem >> 3;
    int j   = (rem & 7) << 2;
    *(v4f*)&B[FRMOFF + (bb * BN + f) * LDST + j] = fr[it];
  }
}

__launch_bounds__(256)
__global__ void stft_wmma_f32(const float* __restrict__ x,
                              const float* __restrict__ cosk,
                              const float* __restrict__ sink,
                              float* __restrict__ out) {
  __shared__ __align__(16) float smem[2 * BUFSZ];    // 55296 B double buffer

  const int tid  = threadIdx.x;
  const int lane = tid & 31;
  const int wave = tid >> 5;
  const int wm = wave & 3;
  const int wn = wave >> 2;

  const int kTile = blockIdx.x * BM;
  const int fTile = blockIdx.y * BN;
  const int bTile = blockIdx.z * BB;
  const bool isEdge = (blockIdx.y == 0) || (blockIdx.y == gridDim.y - 1);

  const int lm    = lane & 15;
  const int lkOff = (lane >> 4) * 2;

  const int offAC = (wm * 16 + lm) * LDST + lkOff;
  const int offAS = SINOFF + offAC;
  const int offB0 = FRMOFF + (wn * 16 + lm) * LDST + lkOff;
  const int offB1 = FRMOFF + (BN + wn * 16 + lm) * LDST + lkOff;

  v8f zero = {};
  v8f accR0 = zero, accI0 = zero, accR1 = zero, accI1 = zero;
  v4f fr[2];

  // ---- prologue: stage chunk 0 into buffer 0
  issue_kern(cosk, sink, smem, tid, kTile, 0);
  if (isEdge) {
    load_frm_edge(x, fr, tid, fTile, bTile, 0);
    store_frm_edge(smem, fr, tid);
  } else {
    issue_frm(x, smem, tid, fTile, bTile, 0);
  }

  for (int c = 0; c < NC; ++c) {
    const int buf  = c & 1;
    const bool more = (c + 1 < NC);
    float* curB = smem + buf * BUFSZ;
    float* nxtB = smem + (buf ^ 1) * BUFSZ;

    // ---- issue next chunk's staging (writes buf^1: safe, last read ended at c-1)
    if (more) {
      const int n0n = (c + 1) * KC;
      issue_kern(cosk, sink, nxtB, tid, kTile, n0n);
      if (isEdge) load_frm_edge(x, fr, tid, fTile, bTile, n0n);
      else        issue_frm(x, nxtB, tid, fTile, bTile, n0n);
    }

    // ---- retire current buffer's async copies (leave next chunk's in flight)
    if (isEdge) { if (more) WAIT_ASYNC(4); else WAIT_ASYNC(0); }
    else        { if (more) WAIT_ASYNC(6); else WAIT_ASYNC(0); }
    __syncthreads();   // buffer `buf` now complete & visible to all waves

    // ---- compute: 8 WMMA k-steps (K=4) on the ready buffer
#pragma unroll
    for (int kk = 0; kk < KC; kk += 4) {
      v2f aC = *(const v2f*)(curB + offAC + kk);
      v2f aS = *(const v2f*)(curB + offAS + kk);
      v2f b0 = *(const v2f*)(curB + offB0 + kk);
      v2f b1 = *(const v2f*)(curB + offB1 + kk);
      accR0 = __builtin_amdgcn_wmma_f32_16x16x4_f32(false, aC, false, b0,
                                                    (short)0, accR0, false, false);
      accI0 = __builtin_amdgcn_wmma_f32_16x16x4_f32(false, aS, false, b0,
                                                    (short)0, accI0, false, false);
      accR1 = __builtin_amdgcn_wmma_f32_16x16x4_f32(false, aC, false, b1,
                                                    (short)0, accR1, false, false);
      accI1 = __builtin_amdgcn_wmma_f32_16x16x4_f32(false, aS, false, b1,
                                                    (short)0, accI1, false, false);
    }
    __syncthreads();   // all waves done reading `buf` before it is overwritten

    if (more && isEdge) store_frm_edge(nxtB, fr, tid);  // overlaps next iter's wait
  }

  // ---- store out[b][k][f][2]
  const int f = fTile + wn * 16 + lm;
  const int kBase = kTile + wm * 16 + ((lane >> 4) << 3);
  if (f < NFRAMES) {
#pragma unroll
    for (int r = 0; r < 8; ++r) {
      int k = kBase + r;
      if (k < NBINS) {
        size_t o0 = (((size_t)(bTile + 0) * NBINS + k) * NFRAMES + f) * 2;
        size_t o1 = (((size_t)(bTile + 1) * NBINS + k) * NFRAMES + f) * 2;
        v2f v0; v0.x = accR0[r]; v0.y = accI0[r];
        v2f v1; v1.x = accR1[r]; v1.y = accI1[r];
        *(v2f*)&out[o0] = v0;
        *(v2f*)&out[o1] = v1;
      }
    }
  }
}

#else  // !HAVE_ASYNC ==========================================================
// Fallback: synchronous single-buffer version (round-1 kernel).
#define KC   64
#define LDST 68

__launch_bounds__(256)
__global__ void stft_wmma_f32(const float* __restrict__ x,
                              const float* __restrict__ cosk,
                              const float* __restrict__ sink,
                              float* __restrict__ out) {
  __shared__ __align__(16) float sCos[BM * LDST];
  __shared__ __align__(16) float sSin[BM * LDST];
  __shared__ __align__(16) float sFrm[BB * BN * LDST];

  const int tid  = threadIdx.x;
  const int lane = tid & 31;
  const int wave = tid >> 5;
  const int wm = wave & 3;
  const int wn = wave >> 2;

  const int kTile = blockIdx.x * BM;
  const int fTile = blockIdx.y * BN;
  const int bTile = blockIdx.z * BB;

  const int lm    = lane & 15;
  const int lkOff = (lane >> 4) * 2;

  v8f zero = {};
  v8f accR[BB], accI[BB];
#pragma unroll
  for (int bb = 0; bb < BB; ++bb) { accR[bb] = zero; accI[bb] = zero; }

  for (int n0 = 0; n0 < N_FFT; n0 += KC) {
#pragma unroll
    for (int it = 0; it < 4; ++it) {
      int i   = tid + it * 256;
      int row = i >> 4;
      int j   = (i & 15) << 2;
      int kIdx = kTile + row;
      kIdx = (kIdx > NBINS - 1) ? (NBINS - 1) : kIdx;
      const size_t goff = (size_t)kIdx * N_FFT + (size_t)(n0 + j);
      v4f c = *(const v4f*)&cosk[goff];
      v4f s = *(const v4f*)&sink[goff];
      *(v4f*)&sCos[row * LDST + j] = c;
      *(v4f*)&sSin[row * LDST + j] = s;
    }
#pragma unroll
    for (int it = 0; it < 4; ++it) {
      int i   = tid + it * 256;
      int bb  = i >> 9;
      int rem = i & 511;
      int f   = rem >> 4;
      int j   = (rem & 15) << 2;
      int fi  = fTile + f;
      fi = (fi > NFRAMES - 1) ? (NFRAMES - 1) : fi;
      int t = fi * HOP + n0 + j - HALF;
      const float* xb = x + (size_t)(bTile + bb) * SIGLEN;
      v4f v;
      if (t >= 0 && t + 3 < SIGLEN) {
        v = *(const v4f*)&xb[t];
      } else {
        v.x = xb[reflect_idx(t)];
        v.y = xb[reflect_idx(t + 1)];
        v.z = xb[reflect_idx(t + 2)];
        v.w = xb[reflect_idx(t + 3)];
      }
      *(v4f*)&sFrm[(bb * BN + f) * LDST + j] = v;
    }
    __syncthreads();

    const float* aCosBase = &sCos[(wm * 16 + lm) * LDST + lkOff];
    const float* aSinBase = &sSin[(wm * 16 + lm) * LDST + lkOff];
    const float* bBase0   = &sFrm[(0 * BN + wn * 16 + lm) * LDST + lkOff];
    const float* bBase1   = &sFrm[(1 * BN + wn * 16 + lm) * LDST + lkOff];
#pragma unroll
    for (int kk = 0; kk < KC; kk += 4) {
      v2f aC = *(const v2f*)(aCosBase + kk);
      v2f aS = *(const v2f*)(aSinBase + kk);
      v2f b0 = *(const v2f*)(bBase0 + kk);
      v2f b1 = *(const v2f*)(bBase1 + kk);
      accR[0] = __builtin_amdgcn_wmma_f32_16x16x4_f32(false, aC, false, b0,
                                                      (short)0, accR[0], false, false);
      accI[0] = __builtin_amdgcn_wmma_f32_16x16x4_f32(false, aS, false, b0,
                                                      (short)0, accI[0], false, false);
      accR[1] = __builtin_amdgcn_wmma_f32_16x16x4_f32(false, aC, false, b1,
                                                      (short)0, accR[1], false, false);
      accI[1] = __builtin_amdgcn_wmma_f32_16x16x4_f32(false, aS, false, b1,
                                                      (short)0, accI[1], false, false);
    }
    __syncthreads();
  }

  const int f = fTile + wn * 16 + lm;
  const int kBase = kTile + wm * 16 + ((lane >> 4) << 3);
  if (f < NFRAMES) {
#pragma unroll
    for (int bb = 0; bb < BB; ++bb) {
      const int b = bTile + bb;
#pragma unroll
      for (int r = 0; r < 8; ++r) {
        int k = kBase + r;
        if (k < NBINS) {
          size_t o = (((size_t)b * NBINS + k) * NFRAMES + f) * 2;
          v2f v; v.x = accR[bb][r]; v.y = accI[bb][r];
          *(v2f*)&out[o] = v;
        }
      }
    }
  }
}
#endif  // HAVE_ASYNC

extern "C" void kernel_launch(void* const* d_in, const int* in_sizes, int n_in,
                              void* d_out, int out_size, void* d_ws, size_t ws_size,
                              hipStream_t stream) {
  (void)in_sizes; (void)n_in; (void)d_ws; (void)ws_size; (void)out_size;
  const float* x    = (const float*)d_in[0];   // [32,1,441000] f32
  const float* cosk = (const float*)d_in[1];   // [2049,4096]  f32
  const float* sink = (const float*)d_in[2];   // [2049,4096]  f32
  float* out = (float*)d_out;                  // [32,2049,431,2] f32

  dim3 grid((NBINS + BM - 1) / BM,             // 33 k-tiles
            (NFRAMES + BN - 1) / BN,           // 14 frame-tiles
            NBATCH / BB);                      // 16 batch-groups
  dim3 block(256);
  stft_wmma_f32<<<grid, block, 0, stream>>>(x, cosk, sink, out);
}